// model_12575664243327
// MI455X (gfx1250) — compile-verified
//
#include <hip/hip_runtime.h>
#include <math.h>

typedef __attribute__((ext_vector_type(2))) float v2f;
typedef __attribute__((ext_vector_type(8))) float v8f;

namespace {

constexpr int NP = 1024;   // primary slots
constexpr int NS = 1024;   // secondary slots
constexpr int D  = 8;      // slot dim
constexpr int R  = 4;      // rules
constexpr int RH = 32;     // rule mlp hidden
constexpr int THREADS = 256;
constexpr int NWAVE   = THREADS / 32;
constexpr int TILES   = NP / 16;
constexpr float TAU = 0.8f;

__device__ inline v8f wmma_f32x4(v2f a, v2f b, v8f c) {
  // D = A(16x4 f32) x B(4x16 f32) + C(16x16 f32)
  return __builtin_amdgcn_wmma_f32_16x16x4_f32(
      /*neg_a=*/false, a, /*neg_b=*/false, b,
      /*c_mod=*/(short)0, c, /*reuse_a=*/false, /*reuse_b=*/false);
}

// Two-layer MLP (8 -> 16 relu -> 16) on a 16-row tile via f32 WMMA.
// xrows: 16 consecutive rows of 8 f32 in global memory.
// Result (16x16, row = slot, col = out feature) left in tbuf (LDS, 256 f32).
__device__ inline void mlp16_tile(const float* __restrict__ xrows,
                                  const float* __restrict__ w1,  // LDS [8][16]
                                  const float* __restrict__ b1,  // LDS [16]
                                  const float* __restrict__ w2,  // LDS [16][16]
                                  const float* __restrict__ b2,  // LDS [16]
                                  float* __restrict__ tbuf,      // LDS 16x16
                                  int lane) {
  const int hf = lane >> 4;   // half-wave: selects K pair
  const int c  = lane & 15;   // row (for A) / column (for B,C,D)
  const float* rp = xrows + c * D;
  // A layout (16x4): lanes0-15 vgpr{0,1} = K{0,1}; lanes16-31 = K{2,3}
  v2f a0 = *(const v2f*)(rp + 2 * hf);        // K = 0..3 chunk
  v2f a1 = *(const v2f*)(rp + 4 + 2 * hf);    // K = 4..7 chunk
  // B layout (4x16): lanes0-15 vgpr{0,1} = rows K{0,1}; lanes16-31 = K{2,3}
  v2f bw0 = { w1[(2 * hf) * 16 + c],     w1[(2 * hf + 1) * 16 + c] };
  v2f bw1 = { w1[(4 + 2 * hf) * 16 + c], w1[(5 + 2 * hf) * 16 + c] };
  v8f acc;
#pragma unroll
  for (int i = 0; i < 8; ++i) acc[i] = b1[c];   // C[m][n] = bias[n]
  acc = wmma_f32x4(a0, bw0, acc);
  acc = wmma_f32x4(a1, bw1, acc);
  // relu, then re-stripe D-layout -> row-major LDS (per-wave buffer)
#pragma unroll
  for (int i = 0; i < 8; ++i) tbuf[(i + 8 * hf) * 16 + c] = fmaxf(acc[i], 0.0f);
  // layer 2: K = 16 -> 4 chained WMMAs; A rows reloaded from LDS (transposed path)
  v8f acc2;
#pragma unroll
  for (int i = 0; i < 8; ++i) acc2[i] = b2[c];
#pragma unroll
  for (int j = 0; j < 4; ++j) {
    v2f aj = *(const v2f*)(&tbuf[c * 16 + 4 * j + 2 * hf]);
    v2f bj = { w2[(4 * j + 2 * hf) * 16 + c], w2[(4 * j + 2 * hf + 1) * 16 + c] };
    acc2 = wmma_f32x4(aj, bj, acc2);
  }
#pragma unroll
  for (int i = 0; i < 8; ++i) tbuf[(i + 8 * hf) * 16 + c] = acc2[i];
}

__global__ __launch_bounds__(THREADS)
void scene_kernel(const float* __restrict__ pdata, const float* __restrict__ sdata,
                  const float* __restrict__ rule_vecs,
                  const float* __restrict__ g1, const float* __restrict__ g2,
                  const float* __restrict__ q_w1, const float* __restrict__ q_b1,
                  const float* __restrict__ q_w2, const float* __restrict__ q_b2,
                  const float* __restrict__ k_w1, const float* __restrict__ k_b1,
                  const float* __restrict__ k_w2, const float* __restrict__ k_b2,
                  const float* __restrict__ qn_w1, const float* __restrict__ qn_b1,
                  const float* __restrict__ qn_w2, const float* __restrict__ qn_b2,
                  const float* __restrict__ kn_w1, const float* __restrict__ kn_b1,
                  const float* __restrict__ kn_w2, const float* __restrict__ kn_b2,
                  const float* __restrict__ rw1, const float* __restrict__ rb1,
                  const float* __restrict__ rw2, const float* __restrict__ rb2,
                  const float* __restrict__ pr_w1, const float* __restrict__ pr_b1,
                  const float* __restrict__ pr_w2, const float* __restrict__ pr_b2,
                  float* __restrict__ out, int B) {
  __shared__ float zbuf[NP * R];            // z over 4096 (stage1) / 1024 (stage3)
  __shared__ float tbuf[NWAVE][256];        // per-wave 16x16 tiles
  __shared__ float w1a[128], b1a[16], w2a[256], b2a[16];   // q MLP
  __shared__ float w1b[128], b1b[16], w2b[256], b2b[16];   // kn MLP
  __shared__ float rkey[64], khid[64];
  __shared__ float redm[THREADS]; __shared__ int redi[THREADS];
  __shared__ float reds[THREADS];
  __shared__ float psq[16], hid16[16];
  __shared__ float s_v; __shared__ int s_is, s_rs;

  const int tid  = threadIdx.x;
  const int lane = tid & 31;
  const int wid  = tid >> 5;
  const int b    = blockIdx.x;
  const float* pd  = pdata + (size_t)b * NP * D;
  const float* sd  = sdata + (size_t)b * NS * D;
  const float* g1b = g1 + (size_t)b * NP * R;
  const float* g2b = g2 + (size_t)b * NS;

  // ---- stage weights + rule_key ----
  for (int i = tid; i < 128; i += THREADS) { w1a[i] = q_w1[i]; w1b[i] = kn_w1[i]; }
  for (int i = tid; i < 256; i += THREADS) { w2a[i] = q_w2[i]; w2b[i] = kn_w2[i]; }
  if (tid < 16) { b1a[tid] = q_b1[tid]; b2a[tid] = q_b2[tid];
                  b1b[tid] = kn_b1[tid]; b2b[tid] = kn_b2[tid]; }
  if (tid < 64) {
    int r = tid >> 4, j = tid & 15;
    float h = k_b1[j];
    for (int k = 0; k < 16; ++k) h += rule_vecs[r * 16 + k] * k_w1[k * 16 + j];
    khid[tid] = fmaxf(h, 0.0f);
  }
  __syncthreads();
  if (tid < 64) {
    int r = tid >> 4, j = tid & 15;
    float o = k_b2[j];
    for (int h = 0; h < 16; ++h) o += khid[r * 16 + h] * k_w2[h * 16 + j];
    rkey[tid] = o;
  }
  __syncthreads();

  // ---- Stage 1: z[n,r] = (sq(pd)[n] . rkey[r] + g1)/tau ----
  for (int t = wid; t < TILES; t += NWAVE) {
    const int base = t * 16;
    mlp16_tile(pd + base * D, w1a, b1a, w2a, b2a, tbuf[wid], lane);
#pragma unroll
    for (int pp = 0; pp < 2; ++pp) {
      int p = lane + 32 * pp;      // 64 (row, r) pairs per tile
      int row = p >> 2, r = p & 3;
      float dot = 0.0f;
      for (int l = 0; l < 16; ++l) dot += tbuf[wid][row * 16 + l] * rkey[r * 16 + l];
      zbuf[(base + row) * 4 + r] = (dot + g1b[(base + row) * 4 + r]) / TAU;
    }
  }
  __syncthreads();
  // argmax (first-index tie-break) + sum-exp over 4096
  {
    float m = -INFINITY; int mi = 0;
    for (int k = 0; k < 16; ++k) {
      int idx = tid * 16 + k; float z = zbuf[idx];
      if (z > m) { m = z; mi = idx; }
    }
    redm[tid] = m; redi[tid] = mi;
    __syncthreads();
    for (int s = THREADS / 2; s > 0; s >>= 1) {
      if (tid < s) {
        float om = redm[tid + s]; int oi = redi[tid + s];
        if (om > redm[tid] || (om == redm[tid] && oi < redi[tid])) { redm[tid] = om; redi[tid] = oi; }
      }
      __syncthreads();
    }
    float gm = redm[0];
    float sl = 0.0f;
    for (int k = 0; k < 16; ++k) sl += expf(zbuf[tid * 16 + k] - gm);
    reds[tid] = sl;
    __syncthreads();
    for (int s = THREADS / 2; s > 0; s >>= 1) {
      if (tid < s) reds[tid] += reds[tid + s];
      __syncthreads();
    }
    if (tid == 0) {
      float y = 1.0f / reds[0];
      s_v  = (1.0f - y) + y;            // straight-through value at argmax
      s_is = redi[0] >> 2;              // primary slot index
      s_rs = redi[0] & 3;               // rule index
    }
    __syncthreads();
  }

  // ---- Stage 2: psq = v * MLP(pd[i*]; qn_*)  (single row) ----
  if (tid < 16) {
    const float* prow = pd + s_is * D;
    float h = qn_b1[tid];
    for (int k = 0; k < D; ++k) h += prow[k] * qn_w1[k * 16 + tid];
    hid16[tid] = fmaxf(h, 0.0f);
  }
  __syncthreads();
  if (tid < 16) {
    float o = qn_b2[tid];
    for (int h = 0; h < 16; ++h) o += hid16[h] * qn_w2[h * 16 + tid];
    psq[tid] = s_v * o;
  }
  __syncthreads();

  // ---- Stage 3: z2[n] = (sk(sd)[n] . psq + g2)/tau ----
  for (int t = wid; t < TILES; t += NWAVE) {
    const int base = t * 16;
    mlp16_tile(sd + base * D, w1b, b1b, w2b, b2b, tbuf[wid], lane);
    if (lane < 16) {
      float dot = 0.0f;
      for (int l = 0; l < 16; ++l) dot += tbuf[wid][lane * 16 + l] * psq[l];
      zbuf[base + lane] = (dot + g2b[base + lane]) / TAU;
    }
  }
  __syncthreads();
  // argmax + sum-exp over 1024
  {
    float m = -INFINITY; int mi = 0;
    for (int k = 0; k < 4; ++k) {
      int idx = tid * 4 + k; float z = zbuf[idx];
      if (z > m) { m = z; mi = idx; }
    }
    redm[tid] = m; redi[tid] = mi;
    __syncthreads();
    for (int s = THREADS / 2; s > 0; s >>= 1) {
      if (tid < s) {
        float om = redm[tid + s]; int oi = redi[tid + s];
        if (om > redm[tid] || (om == redm[tid] && oi < redi[tid])) { redm[tid] = om; redi[tid] = oi; }
      }
      __syncthreads();
    }
    float gm = redm[0];
    float sl = 0.0f;
    for (int k = 0; k < 4; ++k) sl += expf(zbuf[tid * 4 + k] - gm);
    reds[tid] = sl;
    __syncthreads();
    for (int s = THREADS / 2; s > 0; s >>= 1) {
      if (tid < s) reds[tid] += reds[tid + s];
      __syncthreads();
    }
  }

  // ---- Epilogue (tiny): rule MLPs + outputs ----
  if (tid == 0) {
    const float v = s_v; const int is = s_is, rs = s_rs;
    const int js = redi[0];
    const float y2 = 1.0f / reds[0];
    const float v2 = (1.0f - y2) + y2;
    const float* prow = pd + is * D;
    const float* srow = sd + js * D;
    float prim[8], sec[8];
    for (int d = 0; d < 8; ++d) { prim[d] = v * prow[d]; sec[d] = v2 * srow[d]; }
    float rmask[4] = {0.f, 0.f, 0.f, 0.f}; rmask[rs] = v;
    float rin[4] = { prim[0], prim[1], prim[0], prim[1] };
    float ao[4][2];
    for (int r = 0; r < 4; ++r) {
      float o0 = rb2[r * 2 + 0], o1 = rb2[r * 2 + 1];
      for (int t = 0; t < RH; ++t) {
        float h = rb1[r * RH + t];
        for (int i = 0; i < 4; ++i) h += rin[i] * rw1[(r * 4 + i) * RH + t];
        h = fmaxf(h, 0.0f);
        o0 += h * rw2[(r * RH + t) * 2 + 0];
        o1 += h * rw2[(r * RH + t) * 2 + 1];
      }
      ao[r][0] = o0; ao[r][1] = o1;
    }
    float pred0 = v * ao[rs][0], pred1 = v * ao[rs][1];
    float cin[8] = { prim[0], prim[1], sec[0], sec[1],
                     rmask[0], rmask[1], rmask[2], rmask[3] };
    float coll = pr_b2[0];
    for (int t = 0; t < RH; ++t) {
      float h = pr_b1[t];
      for (int i = 0; i < 8; ++i) h += cin[i] * pr_w1[i * RH + t];
      coll += fmaxf(h, 0.0f) * pr_w2[t];
    }
    // outputs concatenated flat in return order
    float* o_pri = out + (size_t)b * 8;
    float* o_sec = out + (size_t)B * 8  + (size_t)b * 8;
    float* o_rm  = out + (size_t)B * 16 + (size_t)b * 4;
    float* o_pr  = out + (size_t)B * 20 + (size_t)b * 2;
    float* o_ao  = out + (size_t)B * 22 + (size_t)b * 8;
    float* o_cl  = out + (size_t)B * 30 + b;
    for (int d = 0; d < 8; ++d) { o_pri[d] = prim[d]; o_sec[d] = sec[d]; }
    for (int r = 0; r < 4; ++r) o_rm[r] = rmask[r];
    o_pr[0] = pred0; o_pr[1] = pred1;
    for (int r = 0; r < 4; ++r) { o_ao[r * 2] = ao[r][0]; o_ao[r * 2 + 1] = ao[r][1]; }
    *o_cl = coll;
  }
}

} // namespace

extern "C" void kernel_launch(void* const* d_in, const int* in_sizes, int n_in,
                              void* d_out, int out_size, void* d_ws, size_t ws_size,
                              hipStream_t stream) {
  (void)d_ws; (void)ws_size; (void)out_size; (void)n_in;
  const float* p[29];
  for (int i = 0; i < 29; ++i) p[i] = (const float*)d_in[i];
  const int B = in_sizes[0] / (NP * D);
  scene_kernel<<<B, THREADS, 0, stream>>>(
      p[0], p[1], p[2], p[3], p[4],
      p[5], p[6], p[7], p[8],
      p[9], p[10], p[11], p[12],
      p[13], p[14], p[15], p[16],
      p[17], p[18], p[19], p[20],
      p[21], p[22], p[23], p[24],
      p[25], p[26], p[27], p[28],
      (float*)d_out, B);
}